// EfficientGlobalPointer_6141803233458
// MI455X (gfx1250) — compile-verified
//
#include <hip/hip_runtime.h>

typedef float v2f __attribute__((ext_vector_type(2)));
typedef float v8f __attribute__((ext_vector_type(8)));

#define HEADS 12
#define HEAD_SIZE 64
#define B_ 2
#define L_ 2048
#define D_ 768
#define HD 128   // 2*HEAD_SIZE
#define NB 24    // 2*HEADS

// workspace layout (float offsets)
#define WS_H    0
#define WS_QW   (B_*L_*HD)                      // 524288
#define WS_KW   (WS_QW + B_*L_*HEAD_SIZE)       // 786432
#define WS_BIAS (WS_KW + B_*L_*HEAD_SIZE)       // 1048576  (+98304 floats)

// ---------------------------------------------------------------------------
// Kernel 1: h = x @ pW + pb   (M = B*L = 4096, N = 128, K = 768)
// one wave per 16x16 output tile; f32 WMMA, K stepped by 4.
// A layout per step kk: lane<16 -> {A[m][4kk], A[m][4kk+1]}, lane>=16 -> {+2,+3}
// B layout mirrors with n = lane%16.
// ---------------------------------------------------------------------------
__global__ __launch_bounds__(32)
void proj_kernel(const float* __restrict__ x, const float* __restrict__ pW,
                 const float* __restrict__ pb, float* __restrict__ h)
{
    const int lane   = threadIdx.x;
    const int lane16 = lane & 15;
    const int half   = lane >> 4;
    const int nt = blockIdx.x;                 // 0..7
    const int mt = blockIdx.y;                 // 0..255 over B*L/16
    const int row = mt * 16 + lane16;          // [0, B*L)
    const int col = nt * 16 + lane16;          // [0, 128)

    v8f c = {};
    const float* xrow = x  + (size_t)row * D_ + 2 * half;
    const float* pcol = pW + (size_t)(2 * half) * HD + col;
    for (int kk = 0; kk < D_ / 4; ++kk) {
        v2f a = *(const v2f*)(xrow + 4 * kk);
        v2f b;
        b.x = pcol[(size_t)(4 * kk) * HD];
        b.y = pcol[(size_t)(4 * kk) * HD + HD];
        c = __builtin_amdgcn_wmma_f32_16x16x4_f32(false, a, false, b,
                                                  (short)0, c, false, false);
    }
    const float bias = pb[col];
    for (int r = 0; r < 8; ++r) {
        const int m = mt * 16 + r + 8 * half;
        h[(size_t)m * HD + col] = c[r] + bias;
    }
}

// ---------------------------------------------------------------------------
// Kernel 2: RoPE(q), RoPE(k) and biasT[b][j][l] = (h[b,l,:]·qW[:,j]+qb[j])/2
// one thread per (b,l); h row stays hot in cache.
// ---------------------------------------------------------------------------
__global__ __launch_bounds__(256)
void rope_bias_kernel(const float* __restrict__ h, const float* __restrict__ qWm,
                      const float* __restrict__ qb,
                      float* __restrict__ qw, float* __restrict__ kw,
                      float* __restrict__ biasT)
{
    const int idx = blockIdx.x * blockDim.x + threadIdx.x;  // 0..B*L-1
    if (idx >= B_ * L_) return;
    const int b = idx / L_;
    const int l = idx - b * L_;
    const float* hr = h + (size_t)idx * HD;

    // bias (tiny GEMV: 128 x 24)
    for (int j = 0; j < NB; ++j) {
        float acc = qb[j];
        for (int cc = 0; cc < HD; ++cc)
            acc += hr[cc] * qWm[(size_t)cc * NB + j];
        biasT[((size_t)b * NB + j) * L_ + l] = acc * 0.5f;
    }

    // RoPE: q[j]=h[2j], k[j]=h[2j+1]; rotation pair (2p,2p+1) uses freq_p
    float* qr = qw + (size_t)idx * HEAD_SIZE;
    float* kr = kw + (size_t)idx * HEAD_SIZE;
    const float lf = (float)l;
    const float LN10000 = 9.210340371976184f;
    for (int p = 0; p < HEAD_SIZE / 2; ++p) {
        const float freq = expf(-((float)(2 * p) / (float)HEAD_SIZE) * LN10000);
        const float ang = lf * freq;
        const float s = sinf(ang), cn = cosf(ang);
        const float q0 = hr[4 * p + 0], q1 = hr[4 * p + 2];
        qr[2 * p + 0] = q0 * cn - q1 * s;
        qr[2 * p + 1] = q1 * cn + q0 * s;
        const float k0 = hr[4 * p + 1], k1 = hr[4 * p + 3];
        kr[2 * p + 0] = k0 * cn - k1 * s;
        kr[2 * p + 1] = k1 * cn + k0 * s;
    }
}

// ---------------------------------------------------------------------------
// Kernel 3: out[b,g,m,n] = (q[b,m]·k[b,n])/8 + biasT[b,2g,n] + biasT[b,2g+1,m]
// masked to upper triangle (n >= m), else -inf.
// one wave per (b, mt, nt) 16x16 tile; S tile computed once, reused by 12 heads.
// ---------------------------------------------------------------------------
__global__ __launch_bounds__(32)
void logits_kernel(const float* __restrict__ qw, const float* __restrict__ kw,
                   const float* __restrict__ biasT, float* __restrict__ out)
{
    const int lane   = threadIdx.x;
    const int lane16 = lane & 15;
    const int half   = lane >> 4;
    const int nt = blockIdx.x;        // 0..127
    const int mt = blockIdx.y;        // 0..127
    const int b  = blockIdx.z;        // 0..1
    const float NEG_INF = -__builtin_inff();
    const int n = nt * 16 + lane16;

    if (mt > nt) {
        // tile entirely below diagonal: pure -inf fill, skip all math
        for (int g = 0; g < HEADS; ++g) {
            float* o = out + (((size_t)(b * HEADS + g) * L_) + mt * 16) * L_ + n;
            for (int r = 0; r < 8; ++r)
                o[(size_t)(r + 8 * half) * L_] = NEG_INF;
        }
        return;
    }

    // S = Q_mt (16x64) x K_nt^T (64x16), f32 WMMA K-stepped by 4
    v8f c = {};
    const float* qrow = qw + ((size_t)(b * L_) + mt * 16 + lane16) * HEAD_SIZE + 2 * half;
    const float* krow = kw + ((size_t)(b * L_) + nt * 16 + lane16) * HEAD_SIZE + 2 * half;
    for (int kk = 0; kk < HEAD_SIZE / 4; ++kk) {
        v2f a  = *(const v2f*)(qrow + 4 * kk);
        v2f bb = *(const v2f*)(krow + 4 * kk);
        c = __builtin_amdgcn_wmma_f32_16x16x4_f32(false, a, false, bb,
                                                  (short)0, c, false, false);
    }
    float s[8];
    for (int r = 0; r < 8; ++r) s[r] = c[r] * 0.125f;   // 1/sqrt(64)

    const bool diag = (mt == nt);
    for (int g = 0; g < HEADS; ++g) {
        const float  bqn = biasT[((size_t)b * NB + 2 * g) * L_ + n];
        const float* bkp = biasT + ((size_t)b * NB + 2 * g + 1) * L_ + mt * 16;
        float* o = out + (((size_t)(b * HEADS + g) * L_) + mt * 16) * L_ + n;
        for (int r = 0; r < 8; ++r) {
            const int mr = r + 8 * half;
            float val = s[r] + bqn + bkp[mr];
            if (diag && lane16 < mr) val = NEG_INF;
            o[(size_t)mr * L_] = val;
        }
    }
}

extern "C" void kernel_launch(void* const* d_in, const int* in_sizes, int n_in,
                              void* d_out, int out_size, void* d_ws, size_t ws_size,
                              hipStream_t stream) {
    const float* x   = (const float*)d_in[0];
    const float* pW  = (const float*)d_in[1];
    const float* pb  = (const float*)d_in[2];
    const float* qWm = (const float*)d_in[3];
    const float* qb  = (const float*)d_in[4];
    float* ws    = (float*)d_ws;
    float* h     = ws + WS_H;
    float* qwb   = ws + WS_QW;
    float* kwb   = ws + WS_KW;
    float* biasT = ws + WS_BIAS;
    float* outp  = (float*)d_out;

    proj_kernel<<<dim3(HD / 16, (B_ * L_) / 16), 32, 0, stream>>>(x, pW, pb, h);
    rope_bias_kernel<<<dim3((B_ * L_) / 256), 256, 0, stream>>>(h, qWm, qb, qwb, kwb, biasT);
    logits_kernel<<<dim3(L_ / 16, L_ / 16, B_), 32, 0, stream>>>(qwb, kwb, biasT, outp);
}